// FCOS_ATSS_Layer_23433341567494
// MI455X (gfx1250) — compile-verified
//
#include <hip/hip_runtime.h>
#include <hip/hip_bf16.h>
#include <stdint.h>

// FCOS/ATSS decode head for MI455X (gfx1250, wave32).
// Memory-bound (~91 MB @ 23.3 TB/s ≈ 4 us). Strategy:
//  - async global->LDS staging of the 80-class logit rows (ASYNCcnt path),
//    contiguous 40 KB tile per block, perfectly coalesced b128 transfers
//  - monotonic-sigmoid trick: reduce raw logits, one sigmoid per position
//  - padded LDS rows (84 floats) -> 16B-aligned async writes, 2-way max conflicts

#define NB   16
#define NH   128
#define NW   128
#define NC   80
#define NPOS (NB * NH * NW)        // 262144
#define POS_PER_BLOCK 128
#define THREADS       128
#define ROW_PAD       84           // floats per padded LDS row (336 B, 16B aligned)

#if defined(__AMDGCN__) && __has_builtin(__builtin_amdgcn_global_load_async_to_lds_b128)
#define USE_ASYNC_LDS 1
#else
#define USE_ASYNC_LDS 0
#endif

#if USE_ASYNC_LDS
// Builtin signature (from hipcc diagnostic): (as(1) int4*, as(3) int4*, imm, imm)
typedef int v4i __attribute__((ext_vector_type(4)));
typedef __attribute__((address_space(1))) v4i* g_v4i_p;
typedef __attribute__((address_space(3))) v4i* l_v4i_p;
__device__ __forceinline__ g_v4i_p as_global_v4(const void* p) {
    return (g_v4i_p)(uintptr_t)p;           // generic global ptr bits == as(1) address
}
__device__ __forceinline__ l_v4i_p as_lds_v4(void* p) {
    return (l_v4i_p)(uint32_t)(uintptr_t)p; // low 32 bits of generic LDS ptr == LDS offset
}
#endif

__device__ __forceinline__ void wait_async_zero() {
#if USE_ASYNC_LDS
#if __has_builtin(__builtin_amdgcn_s_wait_asynccnt)
    __builtin_amdgcn_s_wait_asynccnt(0);
#else
    asm volatile("s_wait_asynccnt 0" ::: "memory");
#endif
#endif
}

__device__ __forceinline__ float fast_sigmoid(float x) {
    return 1.0f / (1.0f + __expf(-x));
}

extern "C" __global__ void __launch_bounds__(THREADS)
fcos_atss_decode_kernel(const float* __restrict__ t_ltrb,
                        const float* __restrict__ conf_logits,
                        const float* __restrict__ cls_logits,
                        const int*   __restrict__ img_h_p,
                        const int*   __restrict__ img_w_p,
                        float*       __restrict__ out)
{
    __shared__ float smem[POS_PER_BLOCK * ROW_PAD];   // 43,008 B

    const int t        = threadIdx.x;
    const int tileBase = blockIdx.x * POS_PER_BLOCK;          // first position of tile
    const float* gtile = cls_logits + (size_t)tileBase * NC;  // contiguous 40 KB region

    // ---- Stage 128 positions x 80 classes into LDS, fully coalesced -------------
    // float4 index f4 = i*128 + t sweeps the tile linearly: global side is a pure
    // contiguous stream; LDS side scatters into 84-float padded rows.
#pragma unroll
    for (int i = 0; i < 20; ++i) {
        const int f4 = i * THREADS + t;      // 0..2559
        const int F  = f4 * 4;               // float index within tile
        const int p  = F / NC;               // position row 0..127
        const int c  = F - p * NC;           // class offset (multiple of 4)
#if USE_ASYNC_LDS
        __builtin_amdgcn_global_load_async_to_lds_b128(
            as_global_v4(gtile + F), as_lds_v4(&smem[p * ROW_PAD + c]), 0, 0);
#else
        const float4 v = *(const float4*)(gtile + F);
        *(float4*)&smem[p * ROW_PAD + c] = v;
#endif
    }
    wait_async_zero();
    __syncthreads();

    // ---- Per-position 80-way max/argmax from LDS (raw logits; sigmoid monotone) --
    const float* row = &smem[t * ROW_PAD];
    float bestV = -3.402823466e38f;
    int   bestI = 0;
#pragma unroll
    for (int c = 0; c < NC; c += 4) {
        const float4 v = *(const float4*)(row + c);
        if (v.x > bestV) { bestV = v.x; bestI = c;     }
        if (v.y > bestV) { bestV = v.y; bestI = c + 1; }
        if (v.z > bestV) { bestV = v.z; bestI = c + 2; }
        if (v.w > bestV) { bestV = v.w; bestI = c + 3; }
    }

    // ---- Box decode ---------------------------------------------------------------
    const int P  = tileBase + t;
    const int hw = P & (NH * NW - 1);     // 16384 = 2^14
    const int h  = hw >> 7;
    const int w  = hw & (NW - 1);

    const float4 lt  = *(const float4*)(t_ltrb + (size_t)P * 4);
    const float conf = conf_logits[P];
    const float imgW = (float)img_w_p[0];
    const float imgH = (float)img_h_p[0];

    const float px = (float)w * 8.0f + 4.0f;   // anchor center x (stride 8)
    const float py = (float)h * 8.0f + 4.0f;   // anchor center y

    const float pl = __expf(lt.x) * 8.0f;
    const float pt = __expf(lt.y) * 8.0f;
    const float pr = __expf(lt.z) * 8.0f;
    const float pb = __expf(lt.w) * 8.0f;

    const float x1 = fminf(fmaxf(px - pl, 0.0f), imgW);
    const float y1 = fminf(fmaxf(py - pt, 0.0f), imgH);
    const float x2 = fminf(fmaxf(px + pr, 0.0f), imgW);
    const float y2 = fminf(fmaxf(py + pb, 0.0f), imgH);

    float4 box;
    box.x = (x1 + x2) * 0.5f;
    box.y = (y1 + y2) * 0.5f;
    box.z = x2 - x1;
    box.w = y2 - y1;

    // ---- Outputs: [p_xywh | cls_idx | confs] concatenated flat --------------------
    *(float4*)(out + (size_t)P * 4) = box;
    out[(size_t)NPOS * 4 + P] = (float)bestI;
    out[(size_t)NPOS * 5 + P] = sqrtf(fast_sigmoid(conf) * fast_sigmoid(bestV));
}

extern "C" void kernel_launch(void* const* d_in, const int* in_sizes, int n_in,
                              void* d_out, int out_size, void* d_ws, size_t ws_size,
                              hipStream_t stream) {
    (void)in_sizes; (void)n_in; (void)d_ws; (void)ws_size; (void)out_size;
    const float* t_ltrb      = (const float*)d_in[0];
    const float* conf_logits = (const float*)d_in[1];
    const float* cls_logits  = (const float*)d_in[2];
    const int*   img_h       = (const int*)d_in[3];
    const int*   img_w       = (const int*)d_in[4];
    float*       out         = (float*)d_out;

    dim3 grid(NPOS / POS_PER_BLOCK);   // 2048 blocks
    dim3 block(THREADS);               // 128 threads = 4 waves (wave32)
    fcos_atss_decode_kernel<<<grid, block, 0, stream>>>(
        t_ltrb, conf_logits, cls_logits, img_h, img_w, out);
}